// emb_LSTM_83090437308833
// MI455X (gfx1250) — compile-verified
//
#include <hip/hip_runtime.h>

typedef __attribute__((ext_vector_type(2))) float v2f;
typedef __attribute__((ext_vector_type(8))) float v8f;

// single v_rcp_f32 instead of the IEEE div_scale/div_fmas/div_fixup sequence
__device__ __forceinline__ float fast_rcp(float z) {
    return __builtin_amdgcn_rcpf(z);
}
__device__ __forceinline__ float fast_sigmoid(float z) {
    return fast_rcp(1.0f + __expf(-z));
}
__device__ __forceinline__ float fast_tanh(float z) {
    // 2/(1+exp(-2z)) - 1
    return __builtin_fmaf(2.0f, fast_rcp(1.0f + __expf(-2.0f * z)), -1.0f);
}

// z[gate,row] = sum_k W[gate,k]*feat[row,k] + bias[gate], feat = [x(10) | h(4) | 0 0]
// computed as 4 chained V_WMMA_F32_16X16X4_F32 per 16-row tile.
__global__ __launch_bounds__(256) void lstm_cell_wmma(
    const float* __restrict__ x, const float* __restrict__ h, const float* __restrict__ c,
    const float* __restrict__ Wix, const float* __restrict__ bix,
    const float* __restrict__ Wfx, const float* __restrict__ bfx,
    const float* __restrict__ Wgx, const float* __restrict__ bgx,
    const float* __restrict__ Wox, const float* __restrict__ box_,
    const float* __restrict__ Wih, const float* __restrict__ bih,
    const float* __restrict__ Wfh, const float* __restrict__ bfh,
    const float* __restrict__ Wgh, const float* __restrict__ bgh,
    const float* __restrict__ Woh, const float* __restrict__ boh,
    float* __restrict__ h_out, float* __restrict__ c_out,
    int nTiles)
{
    const int lane = threadIdx.x & 31;
    const int lo   = lane & 15;       // row-in-tile (B/C/D N index), gate for A (M index)
    const int hi   = lane >> 4;       // half-wave selector
    const int wave = (int)((blockIdx.x * blockDim.x + threadIdx.x) >> 5);
    const int nWaves = (int)((gridDim.x * blockDim.x) >> 5);

    // ---- A operand: per-lane weight slice, loaded once ----
    // A-matrix 16x4 layout: lane m (0-15): K = k0, k0+1 ; lanes 16-31: K = k0+2, k0+3
    const int gate = lo;
    const int grp = gate >> 2, r = gate & 3;
    const float* Wxg = (grp == 0) ? Wix : (grp == 1) ? Wfx : (grp == 2) ? Wgx : Wox;
    const float* Whg = (grp == 0) ? Wih : (grp == 1) ? Wfh : (grp == 2) ? Wgh : Woh;

    v2f A[4];
#pragma unroll
    for (int t = 0; t < 4; ++t) {
        int k0 = 4 * t + 2 * hi;
#pragma unroll
        for (int j = 0; j < 2; ++j) {
            int k = k0 + j;
            float w;
            if (k < 10)      w = Wxg[r * 10 + k];
            else if (k < 14) w = Whg[r * 4 + (k - 10)];
            else             w = 0.0f;
            A[t][j] = w;
        }
    }

    // ---- C accumulator init = fused bias (bx + bh), per D layout: M = v + 8*hi ----
    v8f bias_c;
#pragma unroll
    for (int v = 0; v < 8; ++v) {
        int jgate = v + 8 * hi;
        int jg = jgate >> 2, jr = jgate & 3;
        const float* bxp = (jg == 0) ? bix : (jg == 1) ? bfx : (jg == 2) ? bgx : box_;
        const float* bhp = (jg == 0) ? bih : (jg == 1) ? bfh : (jg == 2) ? bgh : boh;
        bias_c[v] = bxp[jr] + bhp[jr];
    }

    // Branchless activation constants:
    //   lanes 0-15 (hi=0): p0 = sigmoid(z) = 1*rcp(1+exp(-1*z)) + 0
    //   lanes 16-31(hi=1): p0 = tanh(z)    = 2*rcp(1+exp(-2*z)) - 1
    const float act_m   = hi ? 2.0f : 1.0f;
    const float act_nm  = hi ? -2.0f : -1.0f;
    const float act_add = hi ? -1.0f : 0.0f;

    for (int tile = wave; tile < nTiles; tile += nWaves) {
        const int row = (tile << 4) + lo;
        const float* xr = x + (size_t)row * 10;
        const float* hr = h + (size_t)row * 4;

        // ---- B operand: 4x16 feature slice per K-tile ----
        // B layout: VGPR j, lanes 0-15 -> K=j ; lanes 16-31 -> K=j+2  => this lane
        // supplies feat[row, 4t + 2*hi + {0,1}].
        v2f Bv[4];
#pragma unroll
        for (int t = 0; t < 4; ++t) {
            int k0 = 4 * t + 2 * hi;
            v2f bv;
            if (k0 < 10) {                                   // pairs inside x, 8B aligned
                const float2 p = *(const float2*)(xr + k0);
                bv[0] = p.x; bv[1] = p.y;
            } else if (k0 < 14) {                            // pairs inside h
                const float2 p = *(const float2*)(hr + (k0 - 10));
                bv[0] = p.x; bv[1] = p.y;
            } else {                                         // K padding
                bv[0] = 0.0f; bv[1] = 0.0f;
            }
            Bv[t] = bv;
        }

        // ---- 4 chained f32 WMMAs: D = sum_t A_t x B_t + bias ----
        v8f acc = bias_c;
#pragma unroll
        for (int t = 0; t < 4; ++t) {
            acc = __builtin_amdgcn_wmma_f32_16x16x4_f32(
                /*neg_a=*/false, A[t], /*neg_b=*/false, Bv[t],
                /*c_mod=*/(short)0, acc, /*reuse_a=*/false, /*reuse_b=*/false);
        }

        // ---- gating (branchless activations, rcp-based) ----
        // lanes 0-15: acc[0..3]=zi, acc[4..7]=zf ; lanes 16-31: acc[0..3]=zg, acc[4..7]=zo
        float p0[4], p1[4];
#pragma unroll
        for (int j = 0; j < 4; ++j) {
            // p0 = act_m * rcp(1 + exp(act_nm * z)) + act_add  (sigmoid or tanh per half)
            float e0 = __expf(act_nm * acc[j]);
            p0[j] = __builtin_fmaf(act_m, fast_rcp(1.0f + e0), act_add);
            p1[j] = fast_sigmoid(acc[4 + j]);
        }
        float q0[4], q1[4];
#pragma unroll
        for (int j = 0; j < 4; ++j) {
            q0[j] = __shfl_xor(p0[j], 16, 32);
            q1[j] = __shfl_xor(p1[j], 16, 32);
        }

        const float4 cc = *(const float4*)(c + (size_t)row * 4);
        const float cj[4] = {cc.x, cc.y, cc.z, cc.w};

        float cn[4];
#pragma unroll
        for (int j = 0; j < 4; ++j) {
            float iv = hi ? q0[j] : p0[j];
            float fv = hi ? q1[j] : p1[j];
            float gv = hi ? p0[j] : q0[j];
            cn[j] = __builtin_fmaf(fv, cj[j], iv * gv);
        }

        if (hi == 0) {
            // lower half stores c_new for its row
            float4 outv; outv.x = cn[0]; outv.y = cn[1]; outv.z = cn[2]; outv.w = cn[3];
            *(float4*)(c_out + (size_t)row * 4) = outv;
        } else {
            // upper half stores h_new = o * tanh(c_new)
            float4 outv;
            outv.x = p1[0] * fast_tanh(cn[0]);
            outv.y = p1[1] * fast_tanh(cn[1]);
            outv.z = p1[2] * fast_tanh(cn[2]);
            outv.w = p1[3] * fast_tanh(cn[3]);
            *(float4*)(h_out + (size_t)row * 4) = outv;
        }
    }
}

extern "C" void kernel_launch(void* const* d_in, const int* in_sizes, int n_in,
                              void* d_out, int out_size, void* d_ws, size_t ws_size,
                              hipStream_t stream) {
    (void)n_in; (void)d_ws; (void)ws_size; (void)out_size;

    const float* x   = (const float*)d_in[0];
    const float* h   = (const float*)d_in[1];
    const float* c   = (const float*)d_in[2];
    const float* Wix = (const float*)d_in[3];  const float* bix = (const float*)d_in[4];
    const float* Wfx = (const float*)d_in[5];  const float* bfx = (const float*)d_in[6];
    const float* Wgx = (const float*)d_in[7];  const float* bgx = (const float*)d_in[8];
    const float* Wox = (const float*)d_in[9];  const float* box_ = (const float*)d_in[10];
    const float* Wih = (const float*)d_in[11]; const float* bih = (const float*)d_in[12];
    const float* Wfh = (const float*)d_in[13]; const float* bfh = (const float*)d_in[14];
    const float* Wgh = (const float*)d_in[15]; const float* bgh = (const float*)d_in[16];
    const float* Woh = (const float*)d_in[17]; const float* boh = (const float*)d_in[18];

    const int B = in_sizes[0] / 10;          // x is [B, 10]
    float* h_out = (float*)d_out;            // outputs concatenated: h_new then c_new
    float* c_out = h_out + (size_t)B * 4;

    const int nTiles = B / 16;               // 16 rows per wave-tile
    const int threads = 256;                 // 8 wave32 per block
    int blocks = 2048;                       // grid-stride over tiles; ~16K waves
    int maxBlocks = (nTiles * 32 + threads - 1) / threads;
    if (blocks > maxBlocks) blocks = maxBlocks > 0 ? maxBlocks : 1;

    lstm_cell_wmma<<<blocks, threads, 0, stream>>>(
        x, h, c,
        Wix, bix, Wfx, bfx, Wgx, bgx, Wox, box_,
        Wih, bih, Wfh, bfh, Wgh, bgh, Woh, boh,
        h_out, c_out, nTiles);
}